// Voxel_RefinerXL_49744311222583
// MI455X (gfx1250) — compile-verified
//
#include <hip/hip_runtime.h>
#include <math.h>

#define VOL   (128*128*128)
#define PLANE (128*128)

typedef __attribute__((ext_vector_type(2))) float v2f;
typedef __attribute__((ext_vector_type(8))) float v8f;

#define GLOBAL_AS __attribute__((address_space(1)))
#define LDS_AS    __attribute__((address_space(3)))

enum { EPI_RELU = 0, EPI_BIAS = 1, EPI_BIAS1 = 2, EPI_NORM = 3 };

#if defined(__has_builtin)
# if __has_builtin(__builtin_amdgcn_global_load_async_to_lds_b32)
#  define HAVE_ASYNC_LDS 1
# endif
#endif
#ifndef HAVE_ASYNC_LDS
# define HAVE_ASYNC_LDS 0
#endif

// Implicit-GEMM 3x3x3 conv, C_in = 8, via V_WMMA_F32_16X16X4_F32.
// Per wave: M = 16 x-voxels, N = 16*NT output channels, K = 72 rows * 4
// (dx slot 3 is zero in the B panel, so the matching A slot may hold any
//  finite value -> branch-free A fragment loads).
template<int COUT, int NT, int EPI>
__global__ __launch_bounds__(128)
void conv3d_wmma_k(const float* __restrict__ in, const float* __restrict__ w,
                   const float* __restrict__ bias, float* __restrict__ out)
{
    constexpr int LDN = NT * 16;
    __shared__ float Bsm[288 * LDN];          // B fragments: Bsm[k = row*4+dx][n]
    __shared__ float rows[4][72][20];         // per-wave A row cache: x = x0-1 .. x0+17

    const int tid = threadIdx.x;

    // Stage weights zero-padded into the B-matrix LDS panel.
    for (int e = tid; e < 288 * LDN; e += 128) {
        int k = e / LDN, n = e % LDN;
        int r = k >> 2, dx = k & 3;
        float v = 0.f;
        if (dx < 3 && n < COUT) {
            int ci = r / 9, rem = r % 9;
            int dz = rem / 3, dy = rem % 3;
            v = w[((n * 8 + ci) * 27) + dz * 9 + dy * 3 + dx];   // OIDHW
        }
        Bsm[e] = v;
    }

    const int wid  = tid >> 5, lane = tid & 31;
    const int tile = blockIdx.x * 4 + wid;
    const int x0   = (tile & 7) << 4;
    const int y    = (tile >> 3) & 127;
    const int z    = tile >> 10;

    // Stage 72 halo rows of 19 floats (x0-1 .. x0+17), zero-padded at edges.
    // Column 18 is only ever multiplied by the zero dx=3 B-row, but must be
    // a finite value (NaN * 0 = NaN inside WMMA).
    for (int e = lane; e < 72 * 19; e += 32) {
        int r = e / 19, o = e % 19;
        int ci = r / 9, rem = r % 9;
        int zz = z + rem / 3 - 1, yy = y + rem % 3 - 1, xx = x0 + o - 1;
        float v = 0.f;
        if ((unsigned)zz < 128u && (unsigned)yy < 128u && (unsigned)xx < 128u)
            v = in[ci * VOL + zz * PLANE + yy * 128 + xx];
        rows[wid][r][o] = v;
    }
    __syncthreads();

    const int m    = lane & 15;     // A: voxel row / B,D: channel column
    const int half = lane >> 4;
    const int dx0  = half << 1;     // lanes<16 -> K slots k0,k0+1 ; lanes>=16 -> k0+2,k0+3

    v8f acc0 = {0, 0, 0, 0, 0, 0, 0, 0};
    v8f acc1 = {0, 0, 0, 0, 0, 0, 0, 0};

#pragma unroll 4
    for (int j = 0; j < 72; ++j) {
        v2f a;
        a.x = rows[wid][j][m + dx0];
        a.y = rows[wid][j][m + dx0 + 1];      // dx=3 slot: finite * 0 in B
        const float* bk = &Bsm[(j * 4 + dx0) * LDN + m];
        v2f b0; b0.x = bk[0]; b0.y = bk[LDN];
        acc0 = __builtin_amdgcn_wmma_f32_16x16x4_f32(false, a, false, b0,
                                                     (short)0, acc0, false, false);
        if (NT == 2) {                        // second N-tile reuses the A fragment
            v2f b1; b1.x = bk[16]; b1.y = bk[LDN + 16];
            acc1 = __builtin_amdgcn_wmma_f32_16x16x4_f32(false, a, false, b1,
                                                         (short)0, acc1, false, false);
        }
    }

    const int baseIdx = z * PLANE + y * 128 + x0;
#pragma unroll
    for (int r8 = 0; r8 < 8; ++r8) {
        int mv  = r8 + (half << 3);     // D: VGPR r = voxel r (lanes 0-15) / r+8 (lanes 16-31)
        int idx = baseIdx + mv;
        float v0 = acc0[r8];
        if (EPI == EPI_NORM) {
            // L1 norm over 27 channels: |.| reduced across the 16 N-lanes of each half.
            float t = fabsf(v0) + fabsf(acc1[r8]);
            t += __shfl_xor(t, 1);
            t += __shfl_xor(t, 2);
            t += __shfl_xor(t, 4);
            t += __shfl_xor(t, 8);
            float dnm = fmaxf(t, 1e-12f);
            // Weight field (216 MB) is streamed: keep it out of L2 (non-temporal).
            __builtin_nontemporal_store(v0 / dnm, &out[m * VOL + idx]);
            if (m + 16 < 27)
                __builtin_nontemporal_store(acc1[r8] / dnm, &out[(m + 16) * VOL + idx]);
        } else if (EPI == EPI_BIAS1) {
            if (m == 0) out[idx] = v0 + bias[0];
        } else {
            if (m < 8) {
                float v = v0 + bias[m];
                if (EPI == EPI_RELU) v = fmaxf(v, 0.f);
                out[m * VOL + idx] = v;
            }
        }
    }
}

// Data-dependent 27-tap conv, 8 channels. One block per (y,z) row of 128 x-voxels.
// Halo tile staged via async global->LDS copies (ASYNCcnt) when available.
__global__ __launch_bounds__(128)
void adapt8_k(const float* __restrict__ x, const float* __restrict__ wf,
              float* __restrict__ out)
{
    __shared__ float sm[72 * 132];   // 8ch * 3dz * 3dy rows of 130 (x halo), padded stride
    const int tid = threadIdx.x;
    const int y = blockIdx.x & 127, z = blockIdx.x >> 7;

    for (int e = tid; e < 72 * 130; e += 128) {
        int r = e / 130, o = e % 130;
        int c = r / 9, rem = r % 9;
        int zz = z + rem / 3 - 1, yy = y + rem % 3 - 1, xx = o - 1;
        if ((unsigned)zz < 128u && (unsigned)yy < 128u && (unsigned)xx < 128u) {
            const float* src = &x[c * VOL + zz * PLANE + yy * 128 + xx];
#if HAVE_ASYNC_LDS
            __builtin_amdgcn_global_load_async_to_lds_b32(
                (GLOBAL_AS int*)src, (LDS_AS int*)&sm[r * 132 + o], 0, 0);
#else
            sm[r * 132 + o] = *src;
#endif
        } else {
            sm[r * 132 + o] = 0.f;
        }
    }
#if HAVE_ASYNC_LDS
    asm volatile("s_wait_asynccnt 0x0" ::: "memory");
#endif
    __syncthreads();

    const int idx = z * PLANE + y * 128 + tid;
    float wt[27];
#pragma unroll
    for (int t = 0; t < 27; ++t)
        wt[t] = __builtin_nontemporal_load(&wf[t * VOL + idx]);   // stream weight field
    if (z + 2 < 128) __builtin_prefetch(&x[idx + 2 * PLANE], 0, 1);

    float acc[8];
#pragma unroll
    for (int c = 0; c < 8; ++c) acc[c] = 0.f;
#pragma unroll
    for (int c = 0; c < 8; ++c)
#pragma unroll
        for (int ij = 0; ij < 9; ++ij) {
            const float* row = &sm[(c * 9 + ij) * 132 + tid];
            acc[c] += row[0] * wt[ij * 3 + 0] + row[1] * wt[ij * 3 + 1] + row[2] * wt[ij * 3 + 2];
        }
#pragma unroll
    for (int c = 0; c < 8; ++c) out[c * VOL + idx] = acc[c];
}

// 1-channel variant (final stage), optionally fused tanh.
template<bool TANH>
__global__ __launch_bounds__(128)
void adapt1_k(const float* __restrict__ x, const float* __restrict__ wf,
              float* __restrict__ out)
{
    __shared__ float sm[9 * 132];
    const int tid = threadIdx.x;
    const int y = blockIdx.x & 127, z = blockIdx.x >> 7;

    for (int e = tid; e < 9 * 130; e += 128) {
        int r = e / 130, o = e % 130;
        int zz = z + r / 3 - 1, yy = y + r % 3 - 1, xx = o - 1;
        if ((unsigned)zz < 128u && (unsigned)yy < 128u && (unsigned)xx < 128u) {
            const float* src = &x[zz * PLANE + yy * 128 + xx];
#if HAVE_ASYNC_LDS
            __builtin_amdgcn_global_load_async_to_lds_b32(
                (GLOBAL_AS int*)src, (LDS_AS int*)&sm[r * 132 + o], 0, 0);
#else
            sm[r * 132 + o] = *src;
#endif
        } else {
            sm[r * 132 + o] = 0.f;
        }
    }
#if HAVE_ASYNC_LDS
    asm volatile("s_wait_asynccnt 0x0" ::: "memory");
#endif
    __syncthreads();

    const int idx = z * PLANE + y * 128 + tid;
    float acc = 0.f;
#pragma unroll
    for (int ij = 0; ij < 9; ++ij) {
        const float* row = &sm[ij * 132 + tid];
        acc += row[0] * __builtin_nontemporal_load(&wf[(ij * 3 + 0) * VOL + idx])
             + row[1] * __builtin_nontemporal_load(&wf[(ij * 3 + 1) * VOL + idx])
             + row[2] * __builtin_nontemporal_load(&wf[(ij * 3 + 2) * VOL + idx]);
    }
    out[idx] = TANH ? tanhf(acc) : acc;
}

extern "C" void kernel_launch(void* const* d_in, const int* in_sizes, int n_in,
                              void* d_out, int out_size, void* d_ws, size_t ws_size,
                              hipStream_t stream)
{
    (void)in_sizes; (void)n_in; (void)out_size; (void)ws_size;
    const float* x      = (const float*)d_in[0];
    const float* ac1_w1 = (const float*)d_in[1];
    const float* ac1_b1 = (const float*)d_in[2];
    const float* ac1_w2 = (const float*)d_in[3];
    const float* ac2_w1 = (const float*)d_in[4];
    const float* ac2_b1 = (const float*)d_in[5];
    const float* ac2_w2 = (const float*)d_in[6];
    const float* ac3_w1 = (const float*)d_in[7];
    const float* ac3_b1 = (const float*)d_in[8];
    const float* ac3_w2 = (const float*)d_in[9];
    const float* mid_w  = (const float*)d_in[10];
    const float* mid_b  = (const float*)d_in[11];
    const float* out_w  = (const float*)d_in[12];
    const float* out_b  = (const float*)d_in[13];

    float* ws   = (float*)d_ws;
    float* bufA = ws;
    float* bufB = bufA + (size_t)8 * VOL;
    float* bufH = bufB + (size_t)8 * VOL;
    float* wfld = bufH + (size_t)8 * VOL;
    float* fin0 = wfld + (size_t)27 * VOL;
    float* fin1 = fin0 + (size_t)VOL;

    dim3 gC(32768), gA(16384), blk(128);

    // ---- block 1 on x ----
    conv3d_wmma_k<8, 1, EPI_RELU><<<gC, blk, 0, stream>>>(x, ac1_w1, ac1_b1, bufH);
    conv3d_wmma_k<27, 2, EPI_NORM><<<gC, blk, 0, stream>>>(bufH, ac1_w2, nullptr, wfld);
    adapt8_k<<<gA, blk, 0, stream>>>(x,    wfld, bufA);
    adapt8_k<<<gA, blk, 0, stream>>>(bufA, wfld, bufB);
    adapt8_k<<<gA, blk, 0, stream>>>(bufB, wfld, bufA);
    // ---- mid conv ----
    conv3d_wmma_k<8, 1, EPI_BIAS><<<gC, blk, 0, stream>>>(bufA, mid_w, mid_b, bufB);
    // ---- block 2 on mid ----
    conv3d_wmma_k<8, 1, EPI_RELU><<<gC, blk, 0, stream>>>(bufB, ac2_w1, ac2_b1, bufH);
    conv3d_wmma_k<27, 2, EPI_NORM><<<gC, blk, 0, stream>>>(bufH, ac2_w2, nullptr, wfld);
    adapt8_k<<<gA, blk, 0, stream>>>(bufB, wfld, bufA);
    adapt8_k<<<gA, blk, 0, stream>>>(bufA, wfld, bufB);
    adapt8_k<<<gA, blk, 0, stream>>>(bufB, wfld, bufA);        // mid2 in bufA
    // ---- out conv (1 ch) + block 3 (weights from mid2) ----
    conv3d_wmma_k<1, 1, EPI_BIAS1><<<gC, blk, 0, stream>>>(bufA, out_w, out_b, fin0);
    conv3d_wmma_k<8, 1, EPI_RELU><<<gC, blk, 0, stream>>>(bufA, ac3_w1, ac3_b1, bufH);
    conv3d_wmma_k<27, 2, EPI_NORM><<<gC, blk, 0, stream>>>(bufH, ac3_w2, nullptr, wfld);
    adapt1_k<false><<<gA, blk, 0, stream>>>(fin0, wfld, fin1);
    adapt1_k<false><<<gA, blk, 0, stream>>>(fin1, wfld, fin0);
    adapt1_k<true><<<gA, blk, 0, stream>>>(fin0, wfld, (float*)d_out);
}